// NTN_8564164788344
// MI455X (gfx1250) — compile-verified
//
#include <hip/hip_runtime.h>

typedef __attribute__((ext_vector_type(2))) float v2f;
typedef __attribute__((ext_vector_type(8))) float v8f;
typedef __attribute__((ext_vector_type(4))) unsigned int v4u;
typedef __attribute__((ext_vector_type(8))) int v8i;
typedef __attribute__((ext_vector_type(4))) int v4i;

constexpr int NN = 32768;
constexpr int DD = 128;
constexpr int KK = 32;
constexpr float EPS = 1e-8f;

// Issue one TDM descriptor: copy a 128-row x 16-col f32 tile (row stride 128
// elements) from global memory to LDS, packed contiguously as [d*16 + e].
// D# layout per CDNA5 ISA sec 8.3-8.6.
__device__ __forceinline__ void tdm_load_tile(const float* gsrc, unsigned lds_off) {
    unsigned long long ga = (unsigned long long)(uintptr_t)gsrc;
    v4u g0 = { 1u,                                               // count=1, user D#
               lds_off,                                          // lds_addr (bytes)
               (unsigned)(ga & 0xFFFFFFFFu),                     // global_addr[31:0]
               (unsigned)(((ga >> 32) & 0x1FFFFFFu) | (2u << 30)) }; // addr[56:32] | type=2
    v8i g1 = { (int)(2u << 16),    // wg_mask=0, data_size=2 (4B), no pad/iterate
               (int)(128u << 16),  // abar=0 | tensor_dim0[15:0]=128
               (int)(128u << 16),  // tensor_dim0[31:16]=0 | tensor_dim1[15:0]=128
               (int)(16u  << 16),  // tensor_dim1[31:16]=0 | tile_dim0=16
               (int)128,           // tile_dim1=128 | tile_dim2=0
               (int)128,           // tensor_dim0_stride[31:0]=128
               0,                  // stride0[47:32]=0 | dim1_stride[15:0]=0
               0 };                // dim1_stride[47:16]=0
    v4i g2 = { 0, 0, 0, 0 };
    v4i g3 = { 0, 0, 0, 0 };
#if defined(__clang_major__) && (__clang_major__ >= 23)
    v8i g4 = { 0, 0, 0, 0, 0, 0, 0, 0 };
    __builtin_amdgcn_tensor_load_to_lds(g0, g1, g2, g3, g4, 0);
#else
    __builtin_amdgcn_tensor_load_to_lds(g0, g1, g2, g3, 0);
#endif
}

// One workgroup: 8 waves, handles (128-row n-block, one k).
// Wave w owns rows n0..n0+15. e-subtiles t=0..7; W slices double-buffered in
// LDS via TDM (wave 0 issues, overlapped with WMMA on the other buffer).
__global__ __launch_bounds__(256) void ntn_wmma_f32_kernel(
    const float* __restrict__ x1, const float* __restrict__ x2,
    const float* __restrict__ W1, const float* __restrict__ W2,
    const float* __restrict__ V,  const float* __restrict__ bias,
    float* __restrict__ out)
{
    __shared__ float ldsW[2][2][DD * 16];   // [buf][matrix][d*16+e] : 32 KB
    __shared__ float p1s[8][16];            // per-wave part1 staging

    const int tid  = threadIdx.x;
    const int lane = tid & 31;
    const int wid  = tid >> 5;
    const int l16  = lane & 15;
    const int lhal = lane >> 4;

    const int k  = blockIdx.y;
    const int n0 = blockIdx.x * 128 + wid * 16;

    const float* W1k = W1 + (size_t)k * DD * DD;
    const float* W2k = W2 + (size_t)k * DD * DD;

    // Hoist all A fragments (depend only on s, not on t): 64 floats/lane/matrix.
    const float* a1base = x1 + (size_t)(n0 + l16) * DD + lhal * 2;
    const float* a2base = x2 + (size_t)(n0 + l16) * DD + lhal * 2;
    v2f a1f[32], a2f[32];
    #pragma unroll
    for (int s = 0; s < 32; ++s) {
        a1f[s] = *(const v2f*)(a1base + s * 4);
        a2f[s] = *(const v2f*)(a2base + s * 4);
    }

    float dotac[8], s1ac[8], s2ac[8];
    #pragma unroll
    for (int v = 0; v < 8; ++v) { dotac[v] = 0.f; s1ac[v] = 0.f; s2ac[v] = 0.f; }

    // Prologue: TDM-load subtile 0 into buffer 0.
    if (wid == 0) {
        tdm_load_tile(W1k,       (unsigned)(uintptr_t)&ldsW[0][0][0]);
        tdm_load_tile(W2k,       (unsigned)(uintptr_t)&ldsW[0][1][0]);
    }

    for (int t = 0; t < 8; ++t) {
        const int buf = t & 1;
        if (wid == 0) {
            if (t < 7) {
                // Prefetch subtile t+1 into the other buffer, then wait until
                // only those 2 descriptors remain outstanding (t's are done).
                tdm_load_tile(W1k + (t + 1) * 16, (unsigned)(uintptr_t)&ldsW[buf ^ 1][0][0]);
                tdm_load_tile(W2k + (t + 1) * 16, (unsigned)(uintptr_t)&ldsW[buf ^ 1][1][0]);
                __builtin_amdgcn_s_wait_tensorcnt(2);
            } else {
                __builtin_amdgcn_s_wait_tensorcnt(0);
            }
        }
        __syncthreads();

        const float* B1 = &ldsW[buf][0][0];
        const float* B2 = &ldsW[buf][1][0];
        v8f acc1 = {};
        v8f acc2 = {};
        #pragma unroll
        for (int s = 0; s < 32; ++s) {
            int drow = s * 4 + lhal * 2;
            v2f b1 = { B1[drow * 16 + l16], B1[(drow + 1) * 16 + l16] };
            v2f b2 = { B2[drow * 16 + l16], B2[(drow + 1) * 16 + l16] };
            acc1 = __builtin_amdgcn_wmma_f32_16x16x4_f32(
                false, a1f[s], false, b1, (short)0, acc1, false, false);
            acc2 = __builtin_amdgcn_wmma_f32_16x16x4_f32(
                false, a2f[s], false, b2, (short)0, acc2, false, false);
        }
        // Fold this e-subtile into the running cosine-similarity sums.
        #pragma unroll
        for (int v = 0; v < 8; ++v) {
            dotac[v] += acc1[v] * acc2[v];
            s1ac[v]  += acc1[v] * acc1[v];
            s2ac[v]  += acc2[v] * acc2[v];
        }
        __syncthreads();   // all waves done reading buf before TDM rewrites it
    }

    // Reduce over the 16 lanes of each half (the e dimension of the C tile).
    #pragma unroll
    for (int m = 1; m <= 8; m <<= 1) {
        #pragma unroll
        for (int v = 0; v < 8; ++v) {
            dotac[v] += __shfl_xor(dotac[v], m, 32);
            s1ac[v]  += __shfl_xor(s1ac[v],  m, 32);
            s2ac[v]  += __shfl_xor(s2ac[v],  m, 32);
        }
    }

    // part1 for row (lhal*8 + v); stage to LDS to re-map layout (row <- lane).
    if (l16 == 0) {
        #pragma unroll
        for (int v = 0; v < 8; ++v) {
            float n1 = fmaxf(sqrtf(s1ac[v]), EPS);
            float n2 = fmaxf(sqrtf(s2ac[v]), EPS);
            p1s[wid][lhal * 8 + v] = dotac[v] / (n1 * n2);
        }
    }

    // part2: lanes 0-15 do x1[row].V[k][0:128], lanes 16-31 x2[row].V[k][128:256].
    const float* xr = (lhal ? x2 : x1) + (size_t)(n0 + l16) * DD;
    const float* Vk = V + k * 2 * DD + lhal * DD;
    float p2 = 0.f;
    #pragma unroll 8
    for (int j = 0; j < DD; ++j) p2 = fmaf(xr[j], Vk[j], p2);
    p2 += __shfl_xor(p2, 16, 32);

    float res = p1s[wid][l16] + p2 + bias[k];
    if (lane < 16)
        out[(size_t)(n0 + l16) * KK + k] = fmaxf(res, 0.f);
}

extern "C" void kernel_launch(void* const* d_in, const int* in_sizes, int n_in,
                              void* d_out, int out_size, void* d_ws, size_t ws_size,
                              hipStream_t stream) {
    const float* x1 = (const float*)d_in[0];
    const float* x2 = (const float*)d_in[1];
    const float* W1 = (const float*)d_in[2];
    const float* W2 = (const float*)d_in[3];
    const float* V  = (const float*)d_in[4];
    const float* b  = (const float*)d_in[5];
    float* out = (float*)d_out;

    dim3 grid(NN / 128, KK);   // 256 x 32 workgroups
    dim3 block(256);
    ntn_wmma_f32_kernel<<<grid, block, 0, stream>>>(x1, x2, W1, W2, V, b, out);
}